// RadianceField_24266565222572
// MI455X (gfx1250) — compile-verified
//
#include <hip/hip_runtime.h>
#include <hip/hip_bf16.h>
#include <math.h>

// ---------------------------------------------------------------------------
// Radiance-field volumetric renderer for MI455X (gfx1250, wave32).
// One 256-thread workgroup (8 waves) per ray; thread s owns sample s.
//   1. stratified samples + hybrid shuffle/LDS bitonic sort
//      (j<32 strides via v_shfl_xor in registers, only j>=32 touch LDS)
//   2. trilinear gather of 8 corners x (27 SH + opacity) from the 128^3 grid
//      (global_prefetch_b8 on far-plane corner rows)
//   3. SH shading -> per-sample color, opacity
//   4. transmittance cumsum: intra-wave __shfl_up scan + cross-wave offsets
//   5. per-wave compositing reduction on the matrix pipe:
//      8 chained V_WMMA_F32_16X16X4_F32 with A == ones; s_v padded to 4
//      channel rows (row 3 zeroed) so every B ds_load is unconditional.
// ---------------------------------------------------------------------------

#define IDIM 128
#define NCH  27      // 9 SH coeffs * 3 color channels
#define S    256
#define NRAYS 2048

typedef __attribute__((ext_vector_type(2))) float v2f;
typedef __attribute__((ext_vector_type(8))) float v8f;

__global__ __launch_bounds__(256, 2)
void radiance_field_kernel(const float* __restrict__ x,
                           const float* __restrict__ d,
                           const float* __restrict__ tmin,
                           const float* __restrict__ tmax,
                           const float* __restrict__ u,
                           const float* __restrict__ grid,
                           const float* __restrict__ opacity,
                           float* __restrict__ out)
{
    __shared__ float s_samples[S];     // sorted sample t values
    __shared__ float s_v[4 * S];       // coeff*color, channel-major; row 3 = 0
    __shared__ float s_wsum[8];        // per-wave scan sums
    __shared__ float s_partial[8 * 3]; // per-wave WMMA partial sums

    const int tid  = threadIdx.x;      // sample index along the ray
    const int ray  = blockIdx.x;
    const int lane = tid & 31;
    const int wave = tid >> 5;

    // ---- 1. stratified samples -------------------------------------------
    const float t0 = tmin[ray];
    const float t1 = tmax[ray];
    float v = (t1 - t0) * u[ray * S + tid] + t0;

    // ---- hybrid bitonic sort of 256 floats (ascending) -------------------
    for (int k = 2; k <= S; k <<= 1) {
        for (int j = k >> 1; j > 0; j >>= 1) {
            const bool up    = ((tid & k) == 0);
            const bool lower = ((tid & j) == 0);
            float pv;
            if (j >= 32) {                 // cross-wave stride: via LDS
                s_samples[tid] = v;
                __syncthreads();
                pv = s_samples[tid ^ j];
                __syncthreads();
            } else {                       // intra-wave stride: via shuffle
                pv = __shfl_xor(v, j, 32);
            }
            v = (lower == up) ? fminf(v, pv) : fmaxf(v, pv);
        }
    }
    s_samples[tid] = v;                    // sorted values for dts neighbor
    s_v[3 * S + tid] = 0.0f;               // zero WMMA padding row
    __syncthreads();

    // ---- 2. sample point, voxel, trilinear gather ------------------------
    const float t  = v;
    const float ox = x[ray * 3 + 0], oy = x[ray * 3 + 1], oz = x[ray * 3 + 2];
    const float dx = d[ray * 3 + 0], dy = d[ray * 3 + 1], dz = d[ray * 3 + 2];

    const float px = ox + t * dx;
    const float py = oy + t * dy;
    const float pz = oz + t * dz;

    int ix = (int)floorf(px), iy = (int)floorf(py), iz = (int)floorf(pz);
    ix = min(max(ix, 0), IDIM - 2);
    iy = min(max(iy, 0), IDIM - 2);
    iz = min(max(iz, 0), IDIM - 2);

    const float fx = px - (float)ix;
    const float fy = py - (float)iy;
    const float fz = pz - (float)iz;

    const long base = ((long)ix * IDIM + iy) * IDIM + iz;

    // Prefetch far-plane corner rows (consumed at the odd corners, cx = 1).
    __builtin_prefetch(grid + (base + (long)IDIM * IDIM + IDIM + 1) * NCH, 0, 1);
    __builtin_prefetch(grid + (base + (long)IDIM * IDIM) * NCH, 0, 1);

    float acc[NCH];
    float accOp = 0.0f;
#pragma unroll
    for (int c = 0; c < NCH; ++c) acc[c] = 0.0f;

#pragma unroll
    for (int corner = 0; corner < 8; ++corner) {
        const int cx = corner & 1;          // DELTA_IJK = (x, y, z) bit order
        const int cy = (corner >> 1) & 1;
        const int cz = (corner >> 2) & 1;
        const float w = (cx ? fx : 1.0f - fx) *
                        (cy ? fy : 1.0f - fy) *
                        (cz ? fz : 1.0f - fz);
        const long gi = base + (long)cx * (IDIM * IDIM) + (long)cy * IDIM + cz;
        const float* __restrict__ g = grid + gi * NCH;
#pragma unroll
        for (int c = 0; c < NCH; ++c) acc[c] = fmaf(w, g[c], acc[c]);
        accOp = fmaf(w, opacity[gi], accOp);
    }

    // ---- 3. SH shading ---------------------------------------------------
    const float inorm = rsqrtf(dx * dx + dy * dy + dz * dz);
    const float nx = dx * inorm, ny = dy * inorm, nz = dz * inorm;
    float sh9[9];
    sh9[0] = 0.28209479f;
    sh9[1] = 0.48860251f * ny;
    sh9[2] = 0.48860251f * nz;
    sh9[3] = 0.48860251f * nx;
    sh9[4] = 1.09254843f * (nx * ny);
    sh9[5] = 1.09254843f * (ny * nz);
    sh9[6] = 0.31539157f * (3.0f * nz * nz - 1.0f);
    sh9[7] = 1.09254843f * (nx * nz);
    sh9[8] = 0.54627422f * (nx * nx - ny * ny);

    float col[3];
#pragma unroll
    for (int c = 0; c < 3; ++c) {
        float cv = 0.5f;
#pragma unroll
        for (int j = 0; j < 9; ++j) cv = fmaf(acc[c * 9 + j], sh9[j], cv);
        col[c] = fmaxf(cv, 0.0f);
    }

    // ---- 4. dts + exclusive scan (transmittance) -------------------------
    float dts = 0.0f;
    if (tid < S - 1) dts = (s_samples[tid + 1] - t) * accOp;

    float inc = dts;                       // intra-wave inclusive scan
#pragma unroll
    for (int off = 1; off < 32; off <<= 1) {
        const float tmp = __shfl_up(inc, off, 32);
        if (lane >= off) inc += tmp;
    }
    if (lane == 31) s_wsum[wave] = inc;
    __syncthreads();

    float woff = 0.0f;                     // cross-wave exclusive offset
#pragma unroll
    for (int w = 0; w < 7; ++w) woff += (w < wave) ? s_wsum[w] : 0.0f;

    const float cum   = (inc + woff) - dts;               // exclusive prefix
    const float coeff = __expf(-cum) * (1.0f - __expf(-dts)); // 0 at s=S-1

#pragma unroll
    for (int c = 0; c < 3; ++c) s_v[c * S + tid] = coeff * col[c];
    __syncthreads();

    // ---- 5. compositing reduction on the matrix pipe ---------------------
    // Wave w reduces samples [32w, 32w+32). A == 1 so D[m,n] = sum_k B[k,n];
    // lanes with n >= 3 read the zeroed padding row (same-address broadcast).
    const int n    = lane & 15;            // channel column (only 0..2 live)
    const int half = lane >> 4;
    const int cidx = (n < 3) ? n : 3;
    const float* __restrict__ bp = &s_v[cidx * S + wave * 32 + half];

    v8f dacc = {};
    const v2f ones = {1.0f, 1.0f};
#pragma unroll
    for (int chunk = 0; chunk < 8; ++chunk) {
        v2f b;
        b.x = bp[chunk * 4];
        b.y = bp[chunk * 4 + 2];
        dacc = __builtin_amdgcn_wmma_f32_16x16x4_f32(
            false, ones, false, b, (short)0, dacc, false, false);
    }
    // D row M=0 lives in dacc[0], lanes 0..15 = N. Lanes 0..2 hold channels.
    if (lane < 3) s_partial[wave * 3 + lane] = dacc[0];
    __syncthreads();

    if (tid < 3) {
        float sum = 0.0f;
#pragma unroll
        for (int w = 0; w < 8; ++w) sum += s_partial[w * 3 + tid];
        out[ray * 3 + tid] = sum;
    }
}

extern "C" void kernel_launch(void* const* d_in, const int* in_sizes, int n_in,
                              void* d_out, int out_size, void* d_ws, size_t ws_size,
                              hipStream_t stream) {
    const float* x       = (const float*)d_in[0];
    const float* d       = (const float*)d_in[1];
    const float* tmin    = (const float*)d_in[2];
    const float* tmax    = (const float*)d_in[3];
    const float* u       = (const float*)d_in[4];
    const float* grid    = (const float*)d_in[5];
    const float* opacity = (const float*)d_in[6];
    float* out = (float*)d_out;

    radiance_field_kernel<<<NRAYS, S, 0, stream>>>(x, d, tmin, tmax, u,
                                                   grid, opacity, out);
}